// xpainn_23313082483609
// MI455X (gfx1250) — compile-verified
//
#include <hip/hip_runtime.h>
#include <math.h>

typedef float v2f __attribute__((ext_vector_type(2)));
typedef float v8f __attribute__((ext_vector_type(8)));

#define EPSF 1e-8f

// ---------------------------------------------------------------------------
// fp32 WMMA GEMM, register-blocked over CB column tiles:
//   Out[r, c] = act( sum_k A[r,k] * W[k,c] + bias[c] )
// A element (r,k):  A[r*lda + k*acs + blockIdx.z]
// Out element (r,c): Out[r*ldo + c*ocs + blockIdx.z]
// W is (K x Co) row-major. One wave per 16x(16*CB) tile, K%4 == 0.
// All accesses are base-pointer + integer offsets so InferAddressSpaces can
// promote every load to global_load (flat_load would also burn DScnt).
// ---------------------------------------------------------------------------
template<int CB>
__global__ __launch_bounds__(32)
void wmma_gemm(const float* __restrict__ A, int lda, int acs,
               const float* __restrict__ W,
               const float* __restrict__ bias,
               float* __restrict__ Out, int ldo, int ocs,
               int nRows, int K, int Co, int act)
{
    const int lane = threadIdx.x;
    const int col0 = blockIdx.x * (16 * CB);
    const int row0 = blockIdx.y * 16;
    const int z    = blockIdx.z;
    const int hi   = lane >> 4;     // 0 or 1 (lane half -> K sub-pair)
    const int l15  = lane & 15;

    int ar = row0 + l15; if (ar >= nRows) ar = nRows - 1;   // clamp, keep EXEC full

    // element offsets (kept in registers; all loads stay GEPs off A / W)
    size_t aofs = (size_t)ar * lda + (size_t)(hi * 2) * acs + z;
    size_t wofs[CB];
    #pragma unroll
    for (int cb = 0; cb < CB; ++cb) {
        int bc = col0 + cb * 16 + l15; if (bc >= Co) bc = Co - 1;
        wofs[cb] = (size_t)bc + (size_t)(hi * 2) * Co;
    }

    const v8f z8 = {0.f,0.f,0.f,0.f,0.f,0.f,0.f,0.f};
    v8f acc[CB];
    #pragma unroll
    for (int cb = 0; cb < CB; ++cb) acc[cb] = z8;

    const size_t stepA = (size_t)4 * acs;
    const size_t stepB = (size_t)4 * Co;
    const size_t acs_s = (size_t)acs;
    const size_t Co_s  = (size_t)Co;

    for (int k0 = 0; k0 < K; k0 += 4) {
        v2f a;
        a.x = A[aofs];
        a.y = A[aofs + acs_s];
        aofs += stepA;
        __builtin_prefetch(W + wofs[0] + stepB, 0, 1);   // speculative, no branch
        #pragma unroll
        for (int cb = 0; cb < CB; ++cb) {
            v2f b;
            b.x = W[wofs[cb]];
            b.y = W[wofs[cb] + Co_s];
            wofs[cb] += stepB;
            // (neg_a, A, neg_b, B, c_mod, C, reuse_a, reuse_b)
            acc[cb] = __builtin_amdgcn_wmma_f32_16x16x4_f32(
                          false, a, false, b, (short)0, acc[cb], false, false);
        }
    }

    // D layout: VGPR r -> M = r + 8*hi, N = lane&15
    #pragma unroll
    for (int cb = 0; cb < CB; ++cb) {
        const int gcol = col0 + cb * 16 + l15;
        if (gcol < Co) {
            const float bv = bias ? bias[gcol] : 0.f;
            #pragma unroll
            for (int r = 0; r < 8; ++r) {
                const int grow = row0 + r + hi * 8;
                if (grow < nRows) {
                    float v = acc[cb][r] + bv;
                    if (act) v = v / (1.f + __expf(-v));   // SiLU
                    Out[(size_t)grow * ldo + (size_t)gcol * ocs + z] = v;
                }
            }
        }
    }
}

// ---------------------------------------------------------------------------
// Split x_vector (N,480) into v0 (N,128), v1 (N,64,3), v2 (N,32,5)
// ---------------------------------------------------------------------------
__global__ void split_xv(const float* __restrict__ xv,
                         float* __restrict__ v0, float* __restrict__ v1,
                         float* __restrict__ v2, int N)
{
    int idx = blockIdx.x * blockDim.x + threadIdx.x;
    int n = idx / 480, s = idx % 480;
    if (n >= N) return;
    float val = xv[(size_t)n * 480 + s];
    if (s < 128)      v0[(size_t)n * 128 + s]         = val;
    else if (s < 320) v1[(size_t)n * 192 + (s - 128)] = val;
    else              v2[(size_t)n * 160 + (s - 320)] = val;
}

// ---------------------------------------------------------------------------
// Edge message. Per edge e: f[j] = phi[src][j]*(rbf_e . rbf_w[:,j] + rbf_b[j])*env,
// scatter-add into node accumulators at dst. RBF contraction done on the fly
// (never materializes the 737 MB E x 576 W matrix). Per-edge constants are
// staged into LDS with CDNA5 async loads (ASYNCcnt) instead of VGPR round-trip.
// ---------------------------------------------------------------------------
__global__ __launch_bounds__(64)
void edge_message(const float* __restrict__ phi,
                  const float* __restrict__ rbf, const float* __restrict__ rsh,
                  const float* __restrict__ env,
                  const float* __restrict__ rbf_w, const float* __restrict__ rbf_b,
                  const int* __restrict__ esrc, const int* __restrict__ edst,
                  const float* __restrict__ v0, const float* __restrict__ v1,
                  const float* __restrict__ v2,
                  float* __restrict__ xs2, float* __restrict__ v02,
                  float* __restrict__ v12, float* __restrict__ v22, int E)
{
    int e = blockIdx.x;
    if (e >= E) return;
    __shared__ float sdat[32];            // [0..19]=rbf, [20..28]=rsh, [29]=env
    int t = threadIdx.x;
    if (t < 30) {
        const float* g;
        if (t < 20)      g = rbf + (size_t)e * 20 + t;
        else if (t < 29) g = rsh + (size_t)e * 9 + (t - 20);
        else             g = env + e;
        // LDS byte offset = low 32 bits of the generic pointer (LDS aperture).
        unsigned lofs = (unsigned)(unsigned long long)(const void*)&sdat[t];
        asm volatile("global_load_async_to_lds_b32 %0, %1, off"
                     :: "v"(lofs), "v"(g) : "memory");
    }
    asm volatile("s_wait_asynccnt 0x0" ::: "memory");
    __syncthreads();

    const float* srbf = sdat;
    const float* srsh = sdat + 20;
    const float  ev   = sdat[29];

    const int s = esrc[e], d = edst[e];
    const float* ph = phi + (size_t)s * 576;

    auto fcol = [&](int j) -> float {
        float w = rbf_b[j];
        #pragma unroll
        for (int r = 0; r < 20; ++r) w += srbf[r] * rbf_w[r * 576 + j];
        return ph[j] * w * ev;
    };

    // fs -> xs
    for (int j = t; j < 128; j += 64)
        atomicAdd(&xs2[(size_t)d * 128 + j], fcol(j));
    // m0 -> v0
    for (int j = t; j < 128; j += 64) {
        float g1 = fcol(128 + j), g2 = fcol(352 + j);
        atomicAdd(&v02[(size_t)d * 128 + j],
                  g1 * v0[(size_t)s * 128 + j] + g2 * srsh[0]);
    }
    // m1 -> v1 (64 m's, 3 comps)
    {
        int m = t;  // t in [0,64)
        float g1 = fcol(256 + m), g2 = fcol(480 + m);
        #pragma unroll
        for (int i = 0; i < 3; ++i)
            atomicAdd(&v12[(size_t)d * 192 + m * 3 + i],
                      g1 * v1[(size_t)s * 192 + m * 3 + i] + g2 * srsh[1 + i]);
    }
    // m2 -> v2 (32 m's, 5 comps)
    if (t < 32) {
        int m = t;
        float g1 = fcol(320 + m), g2 = fcol(544 + m);
        #pragma unroll
        for (int i = 0; i < 5; ++i)
            atomicAdd(&v22[(size_t)d * 160 + m * 5 + i],
                      g1 * v2[(size_t)s * 160 + m * 5 + i] + g2 * srsh[4 + i]);
    }
}

// ---------------------------------------------------------------------------
// cat = [xs(128), W0(128), ||W1||(64), ||W2||(32)]  -> (N, 352)
// ---------------------------------------------------------------------------
__global__ void build_cat(const float* __restrict__ xs, const float* __restrict__ W0,
                          const float* __restrict__ W1, const float* __restrict__ W2,
                          float* __restrict__ cat, int N)
{
    int idx = blockIdx.x * blockDim.x + threadIdx.x;
    int n = idx / 352, s = idx % 352;
    if (n >= N) return;
    float v;
    if (s < 128)      v = xs[(size_t)n * 128 + s];
    else if (s < 256) v = W0[(size_t)n * 128 + (s - 128)];
    else if (s < 320) {
        const float* p = W1 + (size_t)n * 192 + (s - 256) * 3;
        v = sqrtf(p[0]*p[0] + p[1]*p[1] + p[2]*p[2] + EPSF);
    } else {
        const float* p = W2 + (size_t)n * 160 + (s - 320) * 5;
        float q = 0.f;
        #pragma unroll
        for (int i = 0; i < 5; ++i) q += p[i] * p[i];
        v = sqrtf(q + EPSF);
    }
    cat[(size_t)n * 352 + s] = v;
}

// inner = [U0*W0 (128), sum_i U1*W1 (64), sum_i U2*W2 (32)] -> (N, 224)
__global__ void build_inner(const float* __restrict__ U0, const float* __restrict__ W0,
                            const float* __restrict__ U1, const float* __restrict__ W1,
                            const float* __restrict__ U2, const float* __restrict__ W2,
                            float* __restrict__ inner, int N)
{
    int idx = blockIdx.x * blockDim.x + threadIdx.x;
    int n = idx / 224, s = idx % 224;
    if (n >= N) return;
    float v;
    if (s < 128) v = U0[(size_t)n * 128 + s] * W0[(size_t)n * 128 + s];
    else if (s < 192) {
        int m = s - 128; size_t o = (size_t)n * 192 + m * 3;
        v = U1[o]*W1[o] + U1[o+1]*W1[o+1] + U1[o+2]*W1[o+2];
    } else {
        int m = s - 192; size_t o = (size_t)n * 160 + m * 5;
        float q = 0.f;
        #pragma unroll
        for (int i = 0; i < 5; ++i) q += U2[o+i] * W2[o+i];
        v = q;
    }
    inner[(size_t)n * 224 + s] = v;
}

// xs += a_ss + a_sv*proj ; v0 += a_vv0*U0 ; v1 += a_vv1*U1 ; v2 += a_vv2*U2
// a layout (N,480): [a_ss 0..128 | a_sv 128..256 | a_vv 256..480]
__global__ void apply_update(const float* __restrict__ a, const float* __restrict__ proj,
                             const float* __restrict__ U0, const float* __restrict__ U1,
                             const float* __restrict__ U2,
                             float* __restrict__ xs, float* __restrict__ v0,
                             float* __restrict__ v1, float* __restrict__ v2, int N)
{
    int idx = blockIdx.x * blockDim.x + threadIdx.x;
    int n = idx / 224, s = idx % 224;
    if (n >= N) return;
    const float* an = a + (size_t)n * 480;
    if (s < 128) {
        xs[(size_t)n * 128 + s] += an[s] + an[128 + s] * proj[(size_t)n * 128 + s];
        v0[(size_t)n * 128 + s] += an[256 + s] * U0[(size_t)n * 128 + s];
    } else if (s < 192) {
        int m = s - 128; float av = an[384 + m]; size_t o = (size_t)n * 192 + m * 3;
        #pragma unroll
        for (int i = 0; i < 3; ++i) v1[o + i] += av * U1[o + i];
    } else {
        int m = s - 192; float av = an[448 + m]; size_t o = (size_t)n * 160 + m * 5;
        #pragma unroll
        for (int i = 0; i < 5; ++i) v2[o + i] += av * U2[o + i];
    }
}

// LayerNorm over 128 features, one wave32 per row.
__global__ __launch_bounds__(32)
void ln128(float* __restrict__ x, const float* __restrict__ g,
           const float* __restrict__ b, int N)
{
    int n = blockIdx.x;
    if (n >= N) return;
    int lane = threadIdx.x;
    float* row = x + (size_t)n * 128;
    float v[4]; float s = 0.f, sq = 0.f;
    #pragma unroll
    for (int i = 0; i < 4; ++i) { v[i] = row[lane * 4 + i]; s += v[i]; sq += v[i]*v[i]; }
    for (int off = 16; off; off >>= 1) { s += __shfl_xor(s, off); sq += __shfl_xor(sq, off); }
    float mu  = s  * (1.f / 128.f);
    float var = sq * (1.f / 128.f) - mu * mu;
    float inv = rsqrtf(var + EPSF);
    #pragma unroll
    for (int i = 0; i < 4; ++i) {
        int j = lane * 4 + i;
        row[j] = (v[i] - mu) * inv * g[j] + b[j];
    }
}

// Equivariant RMS norm: v[n,m,:] *= w[m] / sqrt(mean_m(sum_i v^2) + eps)
__global__ __launch_bounds__(32)
void vecln(float* __restrict__ v, const float* __restrict__ w, int N, int M, int I)
{
    int n = blockIdx.x;
    if (n >= N) return;
    int lane = threadIdx.x;
    int stride = M * I;
    float* row = v + (size_t)n * stride;
    float s = 0.f;
    for (int m = lane; m < M; m += 32)
        for (int i = 0; i < I; ++i) { float t = row[m * I + i]; s += t * t; }
    for (int off = 16; off; off >>= 1) s += __shfl_xor(s, off);
    float inv = 1.f / sqrtf(s / (float)M + EPSF);
    for (int m = lane; m < M; m += 32) {
        float sc = w[m] * inv;
        for (int i = 0; i < I; ++i) row[m * I + i] *= sc;
    }
}

// atom_e[n] = h[n,:] . w3 + b3   (one wave per node)
__global__ __launch_bounds__(32)
void atom_energy(const float* __restrict__ h, const float* __restrict__ w3,
                 const float* __restrict__ b3, float* __restrict__ atomE, int N)
{
    int n = blockIdx.x;
    if (n >= N) return;
    int lane = threadIdx.x;
    const float* row = h + (size_t)n * 128;
    float s = 0.f;
    #pragma unroll
    for (int i = 0; i < 4; ++i) { int j = lane * 4 + i; s += row[j] * w3[j]; }
    for (int off = 16; off; off >>= 1) s += __shfl_xor(s, off);
    if (lane == 0) atomE[n] = s + b3[0];
}

__global__ void edge_energy(const float* __restrict__ rbf, const float* __restrict__ env,
                            const float* __restrict__ w, const float* __restrict__ b,
                            const int* __restrict__ ei0, float* __restrict__ atomE, int E)
{
    int e = blockIdx.x * blockDim.x + threadIdx.x;
    if (e >= E) return;
    float s = b[0];
    #pragma unroll
    for (int r = 0; r < 20; ++r) s += rbf[(size_t)e * 20 + r] * w[r];
    atomicAdd(&atomE[ei0[e]], s * env[e]);
}

__global__ void mol_sum(const float* __restrict__ atomE, const int* __restrict__ batch,
                        float* __restrict__ out, int N)
{
    int n = blockIdx.x * blockDim.x + threadIdx.x;
    if (n < N) atomicAdd(&out[batch[n]], atomE[n]);
}

__global__ void zero_f(float* p, int n)
{
    int i = blockIdx.x * blockDim.x + threadIdx.x;
    if (i < n) p[i] = 0.f;
}

// ---------------------------------------------------------------------------
extern "C" void kernel_launch(void* const* d_in, const int* in_sizes, int n_in,
                              void* d_out, int out_size, void* d_ws, size_t ws_size,
                              hipStream_t stream)
{
    const int N = in_sizes[0] / 128;      // 20000
    const int E = in_sizes[4];            // 320000
    const size_t Nn = (size_t)N;

    const float* x_scalar = (const float*)d_in[0];
    const float* x_vector = (const float*)d_in[1];
    const float* node_rbf = (const float*)d_in[2];
    const float* rsh      = (const float*)d_in[3];
    const float* env      = (const float*)d_in[4];
    // params flattened in JAX tree order (dict keys sorted alphabetically)
    const float* eq_beta0  = (const float*)d_in[5];
    const float* eq_gamma0 = (const float*)d_in[6];
    const float* eq_w1     = (const float*)d_in[7];
    const float* eq_w2     = (const float*)d_in[8];
    const float* inv_beta  = (const float*)d_in[9];
    const float* inv_gamma = (const float*)d_in[10];
    // layers at 11 + 18*l : U0,U1,U2,V0,V1,V2,inner_b,inner_w,msg_b1,msg_b2,
    //                       msg_rbf_b,msg_rbf_w,msg_w1,msg_w2,upd_b1,upd_b2,upd_w1,upd_w2
    const float* ro_b1    = (const float*)d_in[65];
    const float* ro_b2    = (const float*)d_in[66];
    const float* ro_b3    = (const float*)d_in[67];
    const float* ro_rbf_b = (const float*)d_in[68];
    const float* ro_rbf_w = (const float*)d_in[69];
    const float* ro_w1    = (const float*)d_in[70];
    const float* ro_w2    = (const float*)d_in[71];
    const float* ro_w3    = (const float*)d_in[72];
    const int* edge_index = (const int*)d_in[73];     // (2,E)
    const int* edge_idx0  = (const int*)d_in[74];
    const int* atom_batch = (const int*)d_in[75];

    float* ws = (float*)d_ws;
    // Workspace layout (floats): two 608N state blocks + scratch (~287 MB total)
    float* stateA = ws;                    // xs|v0|v1|v2 = 128+128+192+160 = 608N
    float* stateB = ws + 608 * Nn;
    float* phi    = ws + 1216 * Nn;        // 576N (reused as 'a' 480N)
    float* h1     = ws + 1792 * Nn;        // 128N
    float* U0     = ws + 1920 * Nn;        // 128N
    float* W0     = ws + 2048 * Nn;        // 128N
    float* U1     = ws + 2176 * Nn;        // 192N
    float* W1     = ws + 2368 * Nn;        // 192N
    float* U2     = ws + 2560 * Nn;        // 160N
    float* W2     = ws + 2720 * Nn;        // 160N
    float* catb   = ws + 2880 * Nn;        // 352N
    float* innerb = ws + 3232 * Nn;        // 224N
    float* proj   = ws + 3456 * Nn;        // 128N (reused as readout h2)
    float* atomE  = ws + 3584 * Nn;        // N
    (void)ws_size; (void)n_in;

    float* cur = stateA;
    float* alt = stateB;

    auto gemm = [&](const float* A, int lda, int acs, const float* W,
                    const float* bias, float* Out, int ldo, int ocs,
                    int rows, int K, int Co, int act, int ic) {
        if (Co % 64 == 0) {
            dim3 g((unsigned)(Co / 64), (unsigned)((rows + 15) / 16), (unsigned)ic);
            wmma_gemm<4><<<g, 32, 0, stream>>>(A, lda, acs, W, bias, Out, ldo, ocs, rows, K, Co, act);
        } else if (Co % 32 == 0) {
            dim3 g((unsigned)(Co / 32), (unsigned)((rows + 15) / 16), (unsigned)ic);
            wmma_gemm<2><<<g, 32, 0, stream>>>(A, lda, acs, W, bias, Out, ldo, ocs, rows, K, Co, act);
        } else {
            dim3 g((unsigned)((Co + 15) / 16), (unsigned)((rows + 15) / 16), (unsigned)ic);
            wmma_gemm<1><<<g, 32, 0, stream>>>(A, lda, acs, W, bias, Out, ldo, ocs, rows, K, Co, act);
        }
    };

    // --- init state ---
    hipMemcpyAsync(cur, x_scalar, Nn * 128 * sizeof(float), hipMemcpyDeviceToDevice, stream);
    split_xv<<<(unsigned)((Nn * 480 + 255) / 256), 256, 0, stream>>>(
        x_vector, cur + 128 * Nn, cur + 256 * Nn, cur + 448 * Nn, N);

    for (int l = 0; l < 3; ++l) {
        const float* const* P = (const float* const*)(d_in + 11 + 18 * l);
        const float* U0w = P[0];  const float* U1w = P[1];
        const float* U2w = P[2];  const float* V0w = P[3];
        const float* V1w = P[4];  const float* V2w = P[5];
        const float* inner_b = P[6];  const float* inner_w = P[7];
        const float* msg_b1  = P[8];  const float* msg_b2  = P[9];
        const float* msg_rbf_b = P[10]; const float* msg_rbf_w = P[11];
        const float* msg_w1  = P[12]; const float* msg_w2  = P[13];
        const float* upd_b1  = P[14]; const float* upd_b2  = P[15];
        const float* upd_w1  = P[16]; const float* upd_w2  = P[17];

        float* xs = cur;              float* v0 = cur + 128 * Nn;
        float* v1 = cur + 256 * Nn;   float* v2 = cur + 448 * Nn;

        // ---- message ----
        gemm(xs, 128, 1, msg_w1, msg_b1, h1, 128, 1, N, 128, 128, /*silu*/1, 1);
        gemm(h1, 128, 1, msg_w2, msg_b2, phi, 576, 1, N, 128, 576, 0, 1);
        hipMemcpyAsync(alt, cur, 608 * Nn * sizeof(float), hipMemcpyDeviceToDevice, stream);
        edge_message<<<(unsigned)E, 64, 0, stream>>>(
            phi, node_rbf, rsh, env, msg_rbf_w, msg_rbf_b,
            edge_index, edge_index + E, v0, v1, v2,
            alt, alt + 128 * Nn, alt + 256 * Nn, alt + 448 * Nn, E);
        { float* t = cur; cur = alt; alt = t; }
        xs = cur; v0 = cur + 128 * Nn; v1 = cur + 256 * Nn; v2 = cur + 448 * Nn;

        // ---- update ----
        gemm(v0, 128, 1, U0w, nullptr, U0, 128, 1, N, 128, 128, 0, 1);
        gemm(v0, 128, 1, V0w, nullptr, W0, 128, 1, N, 128, 128, 0, 1);
        gemm(v1, 192, 3, U1w, nullptr, U1, 192, 3, N, 64, 64, 0, 3);
        gemm(v1, 192, 3, V1w, nullptr, W1, 192, 3, N, 64, 64, 0, 3);
        gemm(v2, 160, 5, U2w, nullptr, U2, 160, 5, N, 32, 32, 0, 5);
        gemm(v2, 160, 5, V2w, nullptr, W2, 160, 5, N, 32, 32, 0, 5);
        build_cat<<<(unsigned)((Nn * 352 + 255) / 256), 256, 0, stream>>>(xs, W0, W1, W2, catb, N);
        gemm(catb, 352, 1, upd_w1, upd_b1, h1, 128, 1, N, 352, 128, 1, 1);
        gemm(h1, 128, 1, upd_w2, upd_b2, phi /* 'a' */, 480, 1, N, 128, 480, 0, 1);
        build_inner<<<(unsigned)((Nn * 224 + 255) / 256), 256, 0, stream>>>(
            U0, W0, U1, W1, U2, W2, innerb, N);
        gemm(innerb, 224, 1, inner_w, inner_b, proj, 128, 1, N, 224, 128, 0, 1);
        apply_update<<<(unsigned)((Nn * 224 + 255) / 256), 256, 0, stream>>>(
            phi, proj, U0, U1, U2, xs, v0, v1, v2, N);

        // ---- norms ----
        ln128<<<(unsigned)N, 32, 0, stream>>>(xs, inv_gamma, inv_beta, N);
        ln128<<<(unsigned)N, 32, 0, stream>>>(v0, eq_gamma0, eq_beta0, N);
        vecln<<<(unsigned)N, 32, 0, stream>>>(v1, eq_w1, N, 64, 3);
        vecln<<<(unsigned)N, 32, 0, stream>>>(v2, eq_w2, N, 32, 5);
    }

    // ---- readout ----
    float* xs = cur;
    gemm(xs, 128, 1, ro_w1, ro_b1, h1, 128, 1, N, 128, 128, 1, 1);
    gemm(h1, 128, 1, ro_w2, ro_b2, proj, 128, 1, N, 128, 128, 1, 1);
    atom_energy<<<(unsigned)N, 32, 0, stream>>>(proj, ro_w3, ro_b3, atomE, N);
    edge_energy<<<(unsigned)((E + 255) / 256), 256, 0, stream>>>(
        node_rbf, env, ro_rbf_w, ro_rbf_b, edge_idx0, atomE, E);
    zero_f<<<(unsigned)((out_size + 255) / 256), 256, 0, stream>>>((float*)d_out, out_size);
    mol_sum<<<(unsigned)((N + 255) / 256), 256, 0, stream>>>(atomE, atom_batch, (float*)d_out, N);
}